// Rule_16767552323843
// MI455X (gfx1250) — compile-verified
//
#include <hip/hip_runtime.h>
#include <stdint.h>

// ---------------- problem constants (fixed by reference) ----------------
#define Hdim   1024
#define Wdim   1024
#define Ndim   (Hdim * Wdim)
#define RKv    2
#define Kv     5
#define KK     (Kv * Kv)          // 25
#define PW     (Wdim + 2 * RKv)   // 1028 padded width
#define PH     (Hdim + 2 * RKv)   // 1028 padded height
#define PLANE  (PW * PH)          // 1,056,784 elements per padded plane
#define TW     32                 // output tile width
#define TH     32                 // output tile height
#define PT     (TW + 2 * RKv)     // 36 = LDS tile pitch (36*4B = 144B, 16B aligned)
#define PTV    (PT / 4)           // 9 float4 per tile row
#define TILE_ELEMS (PT * PT)      // 1296 per plane

typedef __attribute__((ext_vector_type(4))) unsigned int u32x4;
typedef __attribute__((ext_vector_type(8))) int          i32x8;
typedef __attribute__((ext_vector_type(4))) int          i32x4;
typedef __attribute__((ext_vector_type(4))) float        f32x4;

// ---------------------------------------------------------------------------
// Kernel 1: build circularly padded planes in workspace (2D grid, no div/mod).
//   plane0: S0 = spikes(x ch0) * EI
//   plane1: A_old = x ch3
// ---------------------------------------------------------------------------
__global__ __launch_bounds__(256) void pad_planes_kernel(
    const float* __restrict__ x, const float* __restrict__ EI,
    float* __restrict__ pad)
{
    int pj = blockIdx.x * 256 + threadIdx.x;
    int pi = blockIdx.y;
    if (pj >= PW) return;
    int si = pi - RKv; si += (si < 0) ? Hdim : 0; si -= (si >= Hdim) ? Hdim : 0;
    int sj = pj - RKv; sj += (sj < 0) ? Wdim : 0; sj -= (sj >= Wdim) ? Wdim : 0;
    int s   = si * Wdim + sj;
    int idx = pi * PW + pj;
    pad[idx]         = x[s] * EI[s];        // S0
    pad[PLANE + idx] = x[3 * Ndim + s];     // old A
}

// ---------------------------------------------------------------------------
// Kernel 2: fused neuron update + Hebbian weight update.
// Each thread owns 4 consecutive pixels -> all global traffic is B128.
// USE_TDM=true : stage 2x36x36 tile via one Tensor Data Mover 3D load.
// USE_TDM=false: cooperative LDS fill with wrap arithmetic (no workspace).
// ---------------------------------------------------------------------------
template <bool USE_TDM>
__global__ __launch_bounds__(256) void snn_step_kernel(
    const float* __restrict__ pad,
    const float* __restrict__ x,
    const float* __restrict__ EI,
    const float* __restrict__ nnk,
    const float* __restrict__ trm,
    const float* __restrict__ ksI,
    const float* __restrict__ ksE,
    float* __restrict__ out)
{
    __shared__ float tile[2 * TILE_ELEMS];   // [0..1295]=S0, [1296..]=A_old

    const int C0 = blockIdx.x * TW;
    const int R0 = blockIdx.y * TH;

    if (USE_TDM) {
#if __has_builtin(__builtin_amdgcn_tensor_load_to_lds)
        if (threadIdx.x < 32) {
            // ---- Tensor DMA Descriptor (D#), ISA cdna5 ch.8 bit layout ----
            uint64_t gaddr = (uint64_t)(uintptr_t)pad +
                             (uint64_t)((R0 * PW + C0) * 4);
            unsigned lds_base = (unsigned)(uintptr_t)(&tile[0]); // LDS byte offset

            u32x4 g0; i32x8 g1; i32x4 g2, g3; i32x8 g4;
            // group0: count=1 (valid), lds_addr, 57-bit global_addr, type=2
            g0[0] = 1u;
            g0[1] = lds_base;
            g0[2] = (unsigned)(gaddr & 0xFFFFFFFFull);
            g0[3] = (unsigned)((gaddr >> 32) & 0x1FFFFFFull) | (2u << 30);
            // group1: data_size=4B(code2); tensor 1028x1028x2; tile 36x36x2
            g1[0] = 0x00020000;                                   // data_size=2
            g1[1] = (int)((PW & 0xFFFF) << 16);                   // tensor_dim0 lo16
            g1[2] = (int)(((PW >> 16) & 0xFFFF) | ((PH & 0xFFFF) << 16));
            g1[3] = (int)(((PH >> 16) & 0xFFFF) | (PT << 16));    // tile_dim0=36
            g1[4] = (int)(PT | (2 << 16));                        // tile_dim1=36, tile_dim2=2
            g1[5] = (int)PW;                                      // dim0_stride = 1028
            g1[6] = (int)((unsigned)(PLANE & 0xFFFF) << 16);      // dim1_stride lo16
            g1[7] = (int)(PLANE >> 16);                           // dim1_stride hi
            // group2: tensor_dim2 = 2 planes (z-step uses dim1_stride)
            g2[0] = 2; g2[1] = 0; g2[2] = 0; g2[3] = 0;
            g3[0] = 0; g3[1] = 0; g3[2] = 0; g3[3] = 0;
            g4[0] = 0; g4[1] = 0; g4[2] = 0; g4[3] = 0;
            g4[4] = 0; g4[5] = 0; g4[6] = 0; g4[7] = 0;

            __builtin_amdgcn_tensor_load_to_lds(g0, g1, g2, g3, g4, 0);
            __builtin_amdgcn_s_wait_tensorcnt(0);
        }
#endif
        __syncthreads();
    } else {
        for (int t = threadIdx.x; t < TILE_ELEMS; t += 256) {
            int tr = t / PT, tc = t % PT;
            int gr = R0 + tr - RKv; gr += (gr < 0) ? Hdim : 0; gr -= (gr >= Hdim) ? Hdim : 0;
            int gc = C0 + tc - RKv; gc += (gc < 0) ? Wdim : 0; gc -= (gc >= Wdim) ? Wdim : 0;
            int g = gr * Wdim + gc;
            tile[t]              = x[g] * EI[g];
            tile[TILE_ELEMS + t] = x[3 * Ndim + g];
        }
        __syncthreads();
    }

    // thread -> 4 consecutive pixels of one tile row
    const int q  = threadIdx.x & 7;       // quad index in row: cols 4q..4q+3
    const int ly = threadIdx.x >> 3;      // 0..31
    const int p  = (R0 + ly) * Wdim + C0 + 4 * q;

    // 25 per-pixel synaptic weight quads (streamed once -> non-temporal B128)
    f32x4 nn[KK];
#pragma unroll
    for (int k = 0; k < KK; ++k)
        nn[k] = __builtin_nontemporal_load((const f32x4*)&nnk[k * Ndim + p]);

    // input current: 5-row sliding window (8 floats/row) from LDS S0 tile
    f32x4 I = {0.0f, 0.0f, 0.0f, 0.0f};
#pragma unroll
    for (int ky = 0; ky < Kv; ++ky) {
        const f32x4* row = (const f32x4*)&tile[(ly + ky) * PT];
        f32x4 a = row[q], b = row[q + 1];
        float w[8] = {a[0], a[1], a[2], a[3], b[0], b[1], b[2], b[3]};
#pragma unroll
        for (int kx = 0; kx < Kv; ++kx)
#pragma unroll
            for (int j = 0; j < 4; ++j)
                I[j] = fmaf(nn[ky * Kv + kx][j], w[kx + j], I[j]);
    }

    // state loads (all B128)
    f32x4 V  = __builtin_nontemporal_load((const f32x4*)&x[1 * Ndim + p]);
    f32x4 R  = __builtin_nontemporal_load((const f32x4*)&x[2 * Ndim + p]);
    f32x4 T  = __builtin_nontemporal_load((const f32x4*)&x[4 * Ndim + p]);
    f32x4 E  = __builtin_nontemporal_load((const f32x4*)&x[5 * Ndim + p]);
    f32x4 ei = *(const f32x4*)&EI[p];
    f32x4 tr = __builtin_nontemporal_load((const f32x4*)&trm[p]);
    f32x4 kI = __builtin_nontemporal_load((const f32x4*)&ksI[p]);
    f32x4 kE = __builtin_nontemporal_load((const f32x4*)&ksE[p]);

    // old A at the 4 centers (cols 4q+2 .. 4q+5 of A-plane row ly+2)
    f32x4 A;
    {
        const f32x4* arow = (const f32x4*)&tile[TILE_ELEMS + (ly + RKv) * PT];
        f32x4 a = arow[q], b = arow[q + 1];
        float w[8] = {a[0], a[1], a[2], a[3], b[0], b[1], b[2], b[3]};
#pragma unroll
        for (int j = 0; j < 4; ++j) A[j] = w[2 + j];
    }

    f32x4 S, V1, R1, A1, T1, E1, coef;
#pragma unroll
    for (int j = 0; j < 4; ++j) {
        float v = V[j] - 0.2f * V[j] + 4.0f * I[j];
        float s = ((v > T[j]) && (R[j] > 5.0f) && (E[j] > 0.1f)) ? 1.0f : 0.0f;
        S[j]  = s;
        E1[j] = E[j] + 0.0058f * (1.0f - E[j]) - s * 0.1f;
        R1[j] = (R[j] + 1.0f) * (1.0f - s);
        A1[j] = A[j] - A[j] * 0.01f + s;
        T1[j] = T[j] - T[j] * 0.001f + 0.05f * (A1[j] - tr[j]);
        V1[j] = v * (1.0f - s);
        float ei_neg = (ei[j] < 0.0f) ? 1.0f : 0.0f;
        coef[j] = A1[j] - tr[j] * ei_neg;          // post - trm*ei_neg
    }

    // store z = [S, V, R, A, T, E, EI] early to release state registers
    __builtin_nontemporal_store(S,  (f32x4*)&out[0 * Ndim + p]);
    __builtin_nontemporal_store(V1, (f32x4*)&out[1 * Ndim + p]);
    __builtin_nontemporal_store(R1, (f32x4*)&out[2 * Ndim + p]);
    __builtin_nontemporal_store(A1, (f32x4*)&out[3 * Ndim + p]);
    __builtin_nontemporal_store(T1, (f32x4*)&out[4 * Ndim + p]);
    __builtin_nontemporal_store(E1, (f32x4*)&out[5 * Ndim + p]);
    __builtin_nontemporal_store(ei, (f32x4*)&out[6 * Ndim + p]);

    // Hebbian update + clip; accumulate per-pixel sums for renorm
    f32x4 sum = {0.0f, 0.0f, 0.0f, 0.0f};
#pragma unroll
    for (int ky = 0; ky < Kv; ++ky) {
        const f32x4* prow = (const f32x4*)&tile[TILE_ELEMS + (ly + ky) * PT];
        f32x4 a = prow[q], b = prow[q + 1];
        float w[8] = {a[0], a[1], a[2], a[3], b[0], b[1], b[2], b[3]};
#pragma unroll
        for (int kx = 0; kx < Kv; ++kx) {
            const int k = ky * Kv + kx;
#pragma unroll
            for (int j = 0; j < 4; ++j) {
                float mask = (nn[k][j] > 1e-6f) ? 1.0f : 0.0f;
                float nk   = fmaxf(nn[k][j] + 0.01f * w[kx + j] * coef[j] * mask, 0.0f);
                if (k == RKv * Kv + RKv) nk = 0.0f;   // zero center weight
                nn[k][j] = nk;
                sum[j]  += nk;
            }
        }
    }

    // pixel is entirely inhibitory or excitatory -> single renorm scale
    f32x4 scale;
#pragma unroll
    for (int j = 0; j < 4; ++j) {
        float ks = (ei[j] < 0.0f) ? kI[j] : kE[j];
        scale[j] = ks / (sum[j] + 1e-6f);
    }

#pragma unroll
    for (int k = 0; k < KK; ++k)
        __builtin_nontemporal_store(nn[k] * scale, (f32x4*)&out[(7 + k) * Ndim + p]);
}

// ---------------------------------------------------------------------------
extern "C" void kernel_launch(void* const* d_in, const int* in_sizes, int n_in,
                              void* d_out, int out_size, void* d_ws, size_t ws_size,
                              hipStream_t stream)
{
    // setup_inputs order: x, noise_input(unused), EI, nearest_neighbours,
    //                     target_rate_mat, if_inhib(recomputed), k_sums_I, k_sums_E
    const float* x   = (const float*)d_in[0];
    const float* EI  = (const float*)d_in[2];
    const float* nnk = (const float*)d_in[3];
    const float* trm = (const float*)d_in[4];
    const float* ksI = (const float*)d_in[6];
    const float* ksE = (const float*)d_in[7];
    float* out = (float*)d_out;

    dim3 grid(Wdim / TW, Hdim / TH);

    bool tdm_ok = (ws_size >= (size_t)(2 * PLANE) * sizeof(float));
#if !__has_builtin(__builtin_amdgcn_tensor_load_to_lds)
    tdm_ok = false;
#endif

    if (tdm_ok) {
        float* pad = (float*)d_ws;
        dim3 pgrid((PW + 255) / 256, PH);
        pad_planes_kernel<<<pgrid, 256, 0, stream>>>(x, EI, pad);
        snn_step_kernel<true><<<grid, 256, 0, stream>>>(pad, x, EI, nnk, trm,
                                                        ksI, ksE, out);
    } else {
        snn_step_kernel<false><<<grid, 256, 0, stream>>>(nullptr, x, EI, nnk, trm,
                                                         ksI, ksE, out);
    }
}